// FarthestPointSampling_89232240542468
// MI455X (gfx1250) — compile-verified
//
#include <hip/hip_runtime.h>
#include <stdint.h>

#ifndef __has_builtin
#define __has_builtin(x) 0
#endif

// Problem constants (match reference setup_inputs exactly).
#define FPS_B      16
#define FPS_N      131072
#define FPS_NP     2048
#define THREADS    1024
#define NWAVES     (THREADS / 32)            // 32 waves (wave32)
#define PTS_WAVE   (FPS_N / NWAVES)          // 4096 points per wave
#define SUB_PTS    256                       // points per staged sub-chunk
#define NSUB       (PTS_WAVE / SUB_PTS)      // 16 sub-chunks per iteration
#define SUB_BYTES  (SUB_PTS * 12)            // 3072 B per sub-chunk
#define ASY_PER_SUB (SUB_BYTES / (32 * 16))  // 6 b128 async ops per lane
#define BIGD       1e10f

// ---- CDNA5 async global->LDS path (ASYNCcnt), with safe fallbacks ----------
#if __has_builtin(__builtin_amdgcn_global_load_async_to_lds_b128)
#define HAVE_ASYNC_LDS 1
#else
#define HAVE_ASYNC_LDS 0
#endif

#if __has_builtin(__builtin_amdgcn_s_wait_asynccnt)
#define WAIT_ASYNC(n) __builtin_amdgcn_s_wait_asynccnt(n)
#elif HAVE_ASYNC_LDS
#define WAIT_ASYNC(n) asm volatile("s_wait_asynccnt %0" ::"i"(n) : "memory")
#else
#define WAIT_ASYNC(n) ((void)0)
#endif

// Builtin parameter types: (AS1 int4*, AS3 int4*, imm offset, imm cpol)
typedef int v4i __attribute__((ext_vector_type(4)));
typedef __attribute__((address_space(1))) v4i gas_v4i;
typedef __attribute__((address_space(3))) v4i las_v4i;

// Stage one SUB_BYTES chunk of xyz into this wave's LDS buffer.
// Every lane copies 6 x 16B; contiguous 3KB from global, 16B-aligned.
__device__ __forceinline__ void stage_chunk(char* lds_dst, const float* gsrc, int lane) {
#if HAVE_ASYNC_LDS
  char* g = (char*)gsrc;
#pragma unroll
  for (int j = 0; j < ASY_PER_SUB; ++j) {
    __builtin_amdgcn_global_load_async_to_lds_b128(
        (gas_v4i*)(g + lane * 16 + j * 512),
        (las_v4i*)(lds_dst + lane * 16 + j * 512),
        0, 0);
  }
#else
#pragma unroll
  for (int j = 0; j < ASY_PER_SUB; ++j) {
    const float4 v = *(const float4*)((const char*)gsrc + lane * 16 + j * 512);
    *(float4*)(lds_dst + lane * 16 + j * 512) = v;
  }
#endif
}

// One persistent workgroup per batch. 2048 sequential argmax steps.
__global__ __launch_bounds__(THREADS) void FarthestPointSampling_89232240542468_kernel(
    const float* __restrict__ xyz,   // [B, N, 3]
    const int* __restrict__ finit,   // [B]
    float* __restrict__ out,         // [B, NP] (indices as f32, exact for N < 2^24)
    float* __restrict__ dist)        // [B, N] scratch in d_ws
{
  const int b    = blockIdx.x;
  const int tid  = threadIdx.x;
  const int lane = tid & 31;
  const int wave = tid >> 5;

  const float* xb = xyz + (size_t)b * FPS_N * 3;
  float* db       = dist + (size_t)b * FPS_N;
  float* ob       = out + (size_t)b * FPS_NP;

  extern __shared__ char smem[];               // NWAVES * 2 * SUB_BYTES = 192 KB
  __shared__ float redD[NWAVES];
  __shared__ int   redI[NWAVES];
  __shared__ float sC[3];
  __shared__ int   sFar;

  char* wbuf = smem + (size_t)wave * (2 * SUB_BYTES);
  const int waveBase = wave * PTS_WAVE;

  // Init distance = 1e10, strictly on this thread's own slots
  // (same thread later reads/writes them -> no cross-thread global hazards).
#pragma unroll 4
  for (int s = 0; s < PTS_WAVE / 32; ++s)
    db[waveBase + s * 32 + lane] = BIGD;

  if (tid == 0) sFar = finit[b];
  __syncthreads();

  for (int it = 0; it < FPS_NP; ++it) {
    // Prefetch sub-chunk 0 (xyz is centroid-independent) to overlap the barrier.
    stage_chunk(wbuf, xb + (size_t)waveBase * 3, lane);

    if (tid == 0) {
      const int far = sFar;
      ob[it] = (float)far;                      // scan emits the *incoming* farthest
      const float* c = xb + (size_t)far * 3;
      sC[0] = c[0]; sC[1] = c[1]; sC[2] = c[2];
    }
    __syncthreads();                            // centroid visible to all waves
    const float cx = sC[0], cy = sC[1], cz = sC[2];

    float bd = -1.0f;                           // all real distances are >= 0
    int   bi = 0x7fffffff;

    for (int sub = 0; sub < NSUB; ++sub) {
      char* cur = wbuf + (sub & 1) * SUB_BYTES;
      if (sub + 1 < NSUB) {
        stage_chunk(wbuf + ((sub + 1) & 1) * SUB_BYTES,
                    xb + (size_t)(waveBase + (sub + 1) * SUB_PTS) * 3, lane);
        WAIT_ASYNC(ASY_PER_SUB);                // current buffer's 6 ops done
      } else {
        WAIT_ASYNC(0);
      }

      const int gbase = waveBase + sub * SUB_PTS;
#pragma unroll
      for (int s = 0; s < SUB_PTS / 32; ++s) {  // 8 points per lane per sub
        const int g = gbase + s * 32 + lane;
        const float* p = (const float*)(cur + (size_t)(s * 32 + lane) * 12);
        const float dx = p[0] - cx;
        const float dy = p[1] - cy;
        const float dz = p[2] - cz;
        // ((dx*dx + dy*dy) + dz*dz) with contraction blocked to match XLA.
        const float d = __fadd_rn(__fadd_rn(__fmul_rn(dx, dx), __fmul_rn(dy, dy)),
                                  __fmul_rn(dz, dz));
        const float nd = fminf(db[g], d);
        db[g] = nd;
        if (nd > bd || (nd == bd && g < bi)) { bd = nd; bi = g; }
      }
    }

    // Intra-wave argmax (first-max tie-break: prefer lower index).
#pragma unroll
    for (int off = 16; off > 0; off >>= 1) {
      const float od = __shfl_xor(bd, off, 32);
      const int   oi = __shfl_xor(bi, off, 32);
      if (od > bd || (od == bd && oi < bi)) { bd = od; bi = oi; }
    }
    if (lane == 0) { redD[wave] = bd; redI[wave] = bi; }
    __syncthreads();                            // partials visible

    if (wave == 0) {
      float rd = redD[lane];
      int   ri = redI[lane];
#pragma unroll
      for (int off = 16; off > 0; off >>= 1) {
        const float od = __shfl_xor(rd, off, 32);
        const int   oi = __shfl_xor(ri, off, 32);
        if (od > rd || (od == rd && oi < ri)) { rd = od; ri = oi; }
      }
      if (lane == 0) sFar = ri;                 // read next iter by tid 0 only
    }
    // Next iteration's top-of-loop __syncthreads orders redD reuse vs wave0 reads.
  }
}

extern "C" void kernel_launch(void* const* d_in, const int* in_sizes, int n_in,
                              void* d_out, int out_size, void* d_ws, size_t ws_size,
                              hipStream_t stream) {
  const float* xyz   = (const float*)d_in[0];   // [16,131072,3] f32
  const int*   finit = (const int*)d_in[1];     // [16] int
  float*       out   = (float*)d_out;           // [16,2048] indices as f32
  float*       dist  = (float*)d_ws;            // [16,131072] f32 scratch (8 MB)

  const size_t shmem = (size_t)NWAVES * 2 * SUB_BYTES;  // 192 KB (<= 320 KB WG limit)
  FarthestPointSampling_89232240542468_kernel<<<dim3(FPS_B), dim3(THREADS), shmem, stream>>>(
      xyz, finit, out, dist);
}